// REALS_model_3d_62105227100310
// MI455X (gfx1250) — compile-verified
//
#include <hip/hip_runtime.h>
#include <hip/hip_bf16.h>

// Geometry constants from the reference:
//   vol/x_reg logical shape per batch: (D=128, H=128, W=64), flat = (z*128 + h)*64 + w
#define DZ   128
#define HY   128
#define WX   64
#define INPN (DZ * HY * WX)          // 1048576
#define BLK  256
#define BLOCKS_PER_BATCH (INPN / BLK) // 4096

typedef __attribute__((ext_vector_type(2))) float v2f;
typedef __attribute__((ext_vector_type(8))) float v8f;

// -------------------------------------------------------------------------
// Kernel 1: affine grid + trilinear resample (zeros padding, align_corners=F)
// Writes x_reg and a deterministic per-block partial of dot(x_reg[n], w).
// -------------------------------------------------------------------------
__global__ __launch_bounds__(BLK) void resample_dot_kernel(
    const float* __restrict__ x,        // (n, INPN)
    const int*   __restrict__ r_index,  // (n,)
    const float* __restrict__ theta,    // (T, 3, 4)
    const float* __restrict__ w,        // (INPN,)
    float*       __restrict__ x_reg,    // out: (n, INPN)
    float*       __restrict__ partials) // ws: (n, BLOCKS_PER_BATCH)
{
    const int n    = blockIdx.y;
    const int flat = blockIdx.x * BLK + threadIdx.x;

    const int wi =  flat        & (WX - 1);
    const int hi = (flat >> 6)  & (HY - 1);
    const int zi =  flat >> 13;

    // theta row is uniform per block -> scalar loads
    const float* th = theta + (size_t)r_index[n] * 12;

    // align_corners=False normalized coords
    const float xn = (2.0f * wi + 1.0f) * (1.0f / WX) - 1.0f;
    const float yn = (2.0f * hi + 1.0f) * (1.0f / HY) - 1.0f;
    const float zn = (2.0f * zi + 1.0f) * (1.0f / DZ) - 1.0f;

    const float gx = th[0] * xn + th[1] * yn + th[2]  * zn + th[3];
    const float gy = th[4] * xn + th[5] * yn + th[6]  * zn + th[7];
    const float gz = th[8] * xn + th[9] * yn + th[10] * zn + th[11];

    const float ixf = ((gx + 1.0f) * WX - 1.0f) * 0.5f;
    const float iyf = ((gy + 1.0f) * HY - 1.0f) * 0.5f;
    const float izf = ((gz + 1.0f) * DZ - 1.0f) * 0.5f;

    const float fx = floorf(ixf), fy = floorf(iyf), fz = floorf(izf);
    const float tx = ixf - fx,    ty = iyf - fy,    tz = izf - fz;
    const int   x0 = (int)fx,     y0 = (int)fy,     z0 = (int)fz;

    const float* vol = x + (size_t)n * INPN;

    float out = 0.0f;
#pragma unroll
    for (int dz = 0; dz < 2; ++dz) {
#pragma unroll
        for (int dy = 0; dy < 2; ++dy) {
#pragma unroll
            for (int dx = 0; dx < 2; ++dx) {
                const int xc = x0 + dx, yc = y0 + dy, zc = z0 + dz;
                const float wgt = (dx ? tx : 1.0f - tx) *
                                  (dy ? ty : 1.0f - ty) *
                                  (dz ? tz : 1.0f - tz);
                if ((unsigned)xc < (unsigned)WX &&
                    (unsigned)yc < (unsigned)HY &&
                    (unsigned)zc < (unsigned)DZ) {
                    out += wgt * vol[((zc * HY) + yc) * WX + xc];
                }
            }
        }
    }

    x_reg[(size_t)n * INPN + flat] = out;

    // fused partial of dot(x_reg[n], w): deterministic wave32 + LDS tree
    float p = out * w[flat];
#pragma unroll
    for (int off = 16; off > 0; off >>= 1)
        p += __shfl_down(p, off, 32);

    __shared__ float lds[BLK / 32];
    const int lane = threadIdx.x & 31;
    const int wid  = threadIdx.x >> 5;
    if (lane == 0) lds[wid] = p;
    __syncthreads();
    if (threadIdx.x == 0) {
        float s = 0.0f;
#pragma unroll
        for (int i = 0; i < BLK / 32; ++i) s += lds[i];
        partials[(size_t)n * BLOCKS_PER_BATCH + blockIdx.x] = s;
    }
}

// -------------------------------------------------------------------------
// Kernel 2: per-batch reduction of 4096 partials using V_WMMA_F32_16X16X4_F32
// with B == all-ones (D[m][:] += sum_k A[m][k]).  One wave32 per batch,
// EXEC all-ones as WMMA requires.  Deterministic accumulation order.
// Final C/D layout: lanes 0-15 hold rows M=0..7 (VGPR0..7), lanes 16-31 hold
// rows M=8..15; columns are identical, so per-lane 8-sum + shfl_xor(16)
// yields the grand total.
// -------------------------------------------------------------------------
__global__ __launch_bounds__(32) void wmma_reduce_kernel(
    const float* __restrict__ partials,  // (n, BLOCKS_PER_BATCH)
    float*       __restrict__ s_out)     // (n,)
{
    const int n    = blockIdx.x;
    const int lane = threadIdx.x;
    const float* P = partials + (size_t)n * BLOCKS_PER_BATCH;

    v8f acc = {};
    v2f bones;
    bones[0] = 1.0f;
    bones[1] = 1.0f;

    // 64 values consumed per WMMA (2 per lane), 4096/64 = 64 iterations
    for (int it = 0; it < BLOCKS_PER_BATCH / 64; ++it) {
        const float2 a2 = *(const float2*)(P + it * 64 + lane * 2);
        v2f a;
        a[0] = a2.x;
        a[1] = a2.y;
        acc = __builtin_amdgcn_wmma_f32_16x16x4_f32(
            /*neg_a=*/false, a, /*neg_b=*/false, bones,
            /*c_mod=*/(short)0, acc, /*reuse_a=*/false, /*reuse_b=*/false);
    }

    float s = 0.0f;
#pragma unroll
    for (int i = 0; i < 8; ++i) s += acc[i];
    s += __shfl_xor(s, 16, 32);   // add the other half's row-sums

    if (lane == 0) s_out[n] = s;
}

// -------------------------------------------------------------------------
// Kernel 3: L[n, i] = s[n] * w[i]
// -------------------------------------------------------------------------
__global__ __launch_bounds__(BLK) void outer_kernel(
    const float* __restrict__ s,   // (n,)
    const float* __restrict__ w,   // (INPN,)
    float*       __restrict__ L)   // (n, INPN)
{
    const int n = blockIdx.y;
    const int i = blockIdx.x * BLK + threadIdx.x;
    L[(size_t)n * INPN + i] = s[n] * w[i];
}

extern "C" void kernel_launch(void* const* d_in, const int* in_sizes, int n_in,
                              void* d_out, int out_size, void* d_ws, size_t ws_size,
                              hipStream_t stream) {
    const float* x       = (const float*)d_in[0];
    const int*   r_index = (const int*)  d_in[1];
    const float* theta   = (const float*)d_in[2];
    const float* w       = (const float*)d_in[3];
    float*       out     = (float*)d_out;

    const int n = in_sizes[1];  // batch size (8)

    float* partials = (float*)d_ws;                       // n * 4096 floats
    float* s        = partials + (size_t)n * BLOCKS_PER_BATCH; // n floats

    dim3 grid(BLOCKS_PER_BATCH, n);

    resample_dot_kernel<<<grid, BLK, 0, stream>>>(x, r_index, theta, w,
                                                  out, partials);
    wmma_reduce_kernel<<<n, 32, 0, stream>>>(partials, s);
    outer_kernel<<<grid, BLK, 0, stream>>>(s, w, out + (size_t)n * INPN);
}